// GQADecode_43688407335486
// MI455X (gfx1250) — compile-verified
//
#include <hip/hip_runtime.h>
#include <hip/hip_bf16.h>

typedef float v2f __attribute__((ext_vector_type(2)));
typedef float v8f __attribute__((ext_vector_type(8)));
typedef int v4i __attribute__((vector_size(16)));

#define B_ 32
#define H_ 32
#define G_ 8
#define S_ 8192
#define D_ 128
#define HG_ 4
#define NSPLIT 8
#define CHUNK (S_ / NSPLIT)   // 1024 rows per workgroup
#define NW 4                  // waves per workgroup (128 threads)
#define NT_ (NW * 32)         // 128 threads
#define ROWS_W (CHUNK / NW)   // 256 rows per wave
#define NTILE (ROWS_W / 16)   // 16 tiles of 16 rows per wave
#define KPAD 132              // K/Q row pad (dwords): conflict-free b64 reads
#define VPAD 136              // V row pad (dwords): conflict-free b32 reads
#define QSCALE 0.08838834764831845f  // 1/sqrt(128)
#define NEGINF (-3.0e38f)
#define WS_STRIDE (HG_ * 130) // per-partial: [4 heads][128 acc + M + L]

// ---- async global->LDS staging (gfx1250); sync fallback keeps compile safe --
#if defined(__has_builtin)
#if __has_builtin(__builtin_amdgcn_global_load_async_to_lds_b128)
#define HAVE_ASYNC_LDS 1
#endif
#endif
#ifndef HAVE_ASYNC_LDS
#define HAVE_ASYNC_LDS 0
#endif

#if HAVE_ASYNC_LDS
__device__ __forceinline__ void copy16B(const float* g, float* l) {
  __builtin_amdgcn_global_load_async_to_lds_b128(
      (__attribute__((address_space(1))) v4i*)g,
      (__attribute__((address_space(3))) v4i*)l, 0, 0);
}
#define WAIT_ASYNC_PREV() asm volatile("s_wait_asynccnt 32" ::: "memory")
#define WAIT_ASYNC_ALL()  asm volatile("s_wait_asynccnt 0" ::: "memory")
#else
__device__ __forceinline__ void copy16B(const float* g, float* l) {
  const float4 t = *reinterpret_cast<const float4*>(g);
  *reinterpret_cast<float4*>(l) = t;
}
#define WAIT_ASYNC_PREV() asm volatile("" ::: "memory")
#define WAIT_ASYNC_ALL()  asm volatile("" ::: "memory")
#endif

// Stage one 16 x 128 f32 tile: instruction r copies row r (512 B, all 32
// lanes, 16 B each) -> perfectly coalesced; LDS rows padded to PAD dwords.
template <int PAD>
__device__ __forceinline__ void stageTile(const float* __restrict__ gRow0,
                                          size_t rowStrideF,
                                          float* __restrict__ lbuf, int lane) {
#pragma unroll
  for (int r = 0; r < 16; ++r)
    copy16B(gRow0 + (size_t)r * rowStrideF + lane * 4, lbuf + r * PAD + lane * 4);
}

__device__ __forceinline__ float groupMax16(float v) {
  v = fmaxf(v, __shfl_xor(v, 8, 16));
  v = fmaxf(v, __shfl_xor(v, 4, 16));
  v = fmaxf(v, __shfl_xor(v, 2, 16));
  v = fmaxf(v, __shfl_xor(v, 1, 16));
  return v;
}
__device__ __forceinline__ float groupSum16(float v) {
  v += __shfl_xor(v, 8, 16);
  v += __shfl_xor(v, 4, 16);
  v += __shfl_xor(v, 2, 16);
  v += __shfl_xor(v, 1, 16);
  return v;
}

// ---------------------------------------------------------------------------
// Kernel 1: flash-decode partials. grid = B*G*NSPLIT, block = 128 (4 waves).
// Per wave: 16-row KV tiles double-buffered in LDS via async copies,
// scores + P.V through V_WMMA_F32_16X16X4_F32, online softmax per head.
// ---------------------------------------------------------------------------
__global__ __launch_bounds__(NT_) void gqa_partial(
    const float* __restrict__ Q, const float* __restrict__ K,
    const float* __restrict__ V, const unsigned char* __restrict__ mask,
    float* __restrict__ ws)
{
  __shared__ float Qs[16][KPAD];            // zero-padded, pre-scaled Q
  __shared__ float Kt[NW][2][16][KPAD];     // double-buffered K tiles
  __shared__ float Vt[NW][2][16][VPAD];     // double-buffered V tiles
  __shared__ float Pbuf[NW][16][20];        // per-wave P tile (head x kvrow)
  __shared__ float accS[NW][HG_][D_];       // wave partial outputs
  __shared__ float mlS[NW][HG_][2];         // wave (M, L)

  const int tid  = threadIdx.x;
  const int wave = tid >> 5;
  const int lane = tid & 31;
  const int rl   = lane & 15;
  const int hi   = lane >> 4;

  const int wg    = blockIdx.x;
  const int split = wg % NSPLIT;
  const int bg    = wg / NSPLIT;
  const int g     = bg % G_;
  const int b     = bg / G_;

  for (int i = tid; i < 16 * KPAD; i += NT_) (&Qs[0][0])[i] = 0.0f;
  for (int i = tid; i < NW * 16 * 20; i += NT_) (&Pbuf[0][0][0])[i] = 0.0f;
  __syncthreads();
  for (int i = tid; i < HG_ * D_; i += NT_) {
    const int h = i >> 7, d = i & 127;
    Qs[h][d] = Q[((size_t)b * H_ + g * HG_ + h) * D_ + d] * QSCALE;
  }
  __syncthreads();

  const size_t rowStride = (size_t)G_ * D_;  // floats between KV rows
  const float* Kg = K + (size_t)b * S_ * rowStride + (size_t)g * D_;
  const float* Vg = V + (size_t)b * S_ * rowStride + (size_t)g * D_;
  const unsigned char* mrow = mask + (size_t)b * S_;

  float M[HG_], L[HG_];
#pragma unroll
  for (int h = 0; h < HG_; ++h) { M[h] = -1.0e30f; L[h] = 0.0f; }
  v8f acc[8];
#pragma unroll
  for (int db = 0; db < 8; ++db) acc[db] = (v8f){0,0,0,0,0,0,0,0};

  const int sbase = split * CHUNK + wave * ROWS_W;

  // prologue: stage tile 0
  stageTile<KPAD>(Kg + (size_t)sbase * rowStride, rowStride,
                  &Kt[wave][0][0][0], lane);
  stageTile<VPAD>(Vg + (size_t)sbase * rowStride, rowStride,
                  &Vt[wave][0][0][0], lane);

  for (int t = 0; t < NTILE; ++t) {
    const int cur = t & 1;
    if (t + 1 < NTILE) {
      const int s1 = sbase + (t + 1) * 16;
      stageTile<KPAD>(Kg + (size_t)s1 * rowStride, rowStride,
                      &Kt[wave][cur ^ 1][0][0], lane);
      stageTile<VPAD>(Vg + (size_t)s1 * rowStride, rowStride,
                      &Vt[wave][cur ^ 1][0][0], lane);
      WAIT_ASYNC_PREV();   // retire tile t's 32 copies; keep t+1 in flight
    } else {
      WAIT_ASYNC_ALL();
    }
    const float* Kb = &Kt[wave][cur][0][0];
    const float* Vb = &Vt[wave][cur][0][0];
    const int s0 = sbase + t * 16;

    // ---- scores tile: C(16x16) = Qpad(16x128) x Ktile^T(128x16) ----
    v8f c0 = (v8f){0,0,0,0,0,0,0,0};
    v8f c1 = (v8f){0,0,0,0,0,0,0,0};
#pragma unroll
    for (int kk = 0; kk < 32; kk += 2) {
      {
        const int off = 4 * kk + 2 * hi;
        const float2 qa = *reinterpret_cast<const float2*>(&Qs[rl][off]);
        const float2 kb = *reinterpret_cast<const float2*>(&Kb[rl * KPAD + off]);
        v2f a; a.x = qa.x; a.y = qa.y;
        v2f bb; bb.x = kb.x; bb.y = kb.y;
        c0 = __builtin_amdgcn_wmma_f32_16x16x4_f32(false, a, false, bb,
                                                   (short)0, c0, false, false);
      }
      {
        const int off = 4 * (kk + 1) + 2 * hi;
        const float2 qa = *reinterpret_cast<const float2*>(&Qs[rl][off]);
        const float2 kb = *reinterpret_cast<const float2*>(&Kb[rl * KPAD + off]);
        v2f a; a.x = qa.x; a.y = qa.y;
        v2f bb; bb.x = kb.x; bb.y = kb.y;
        c1 = __builtin_amdgcn_wmma_f32_16x16x4_f32(false, a, false, bb,
                                                   (short)0, c1, false, false);
      }
    }
    const v8f c = c0 + c1;

    // ---- online softmax across the 16 KV rows (lanes 0-15 per head) ----
    const unsigned char mv = mrow[s0 + rl];
    float corr[HG_], pv[HG_];
#pragma unroll
    for (int h = 0; h < HG_; ++h) {
      const float sc = mv ? c[h] : NEGINF;
      const float tm = groupMax16(sc);
      const float nm = fmaxf(M[h], tm);
      const float cr = __expf(M[h] - nm);
      const float p  = __expf(sc - nm);
      const float ps = groupSum16(p);
      L[h] = L[h] * cr + ps;
      M[h] = nm;
      corr[h] = cr;
      pv[h] = p;
    }
    if (lane < 16) {
#pragma unroll
      for (int h = 0; h < HG_; ++h) Pbuf[wave][h][lane] = pv[h];
    }
#pragma unroll
    for (int db = 0; db < 8; ++db) {
#pragma unroll
      for (int j = 0; j < HG_; ++j) acc[db][j] *= corr[j];
    }
    asm volatile("s_wait_dscnt 0" ::: "memory");

    // ---- out tile: acc(16x128) += P(16x16) x Vtile(16x128) ----
#pragma unroll
    for (int kk = 0; kk < 4; ++kk) {
      const int r = 4 * kk + 2 * hi;
      const float2 pa = *reinterpret_cast<const float2*>(&Pbuf[wave][rl][r]);
      v2f a; a.x = pa.x; a.y = pa.y;
#pragma unroll
      for (int db = 0; db < 8; ++db) {
        v2f bb;
        bb.x = Vb[r * VPAD + db * 16 + rl];
        bb.y = Vb[(r + 1) * VPAD + db * 16 + rl];
        acc[db] = __builtin_amdgcn_wmma_f32_16x16x4_f32(false, a, false, bb,
                                                        (short)0, acc[db],
                                                        false, false);
      }
    }
  }

  // ---- spill wave partials to LDS ----
  if (lane < 16) {
#pragma unroll
    for (int db = 0; db < 8; ++db)
#pragma unroll
      for (int h = 0; h < HG_; ++h)
        accS[wave][h][db * 16 + lane] = acc[db][h];
  }
  if (lane == 0) {
#pragma unroll
    for (int h = 0; h < HG_; ++h) {
      mlS[wave][h][0] = M[h];
      mlS[wave][h][1] = L[h];
    }
  }
  __syncthreads();

  // ---- merge wave partials (log-sum-exp), emit workgroup partial ----
  float* wsp = ws + (size_t)wg * WS_STRIDE;
  for (int i = tid; i < HG_ * D_; i += NT_) {
    const int h = i >> 7, d = i & 127;
    float mg = -1.0e30f;
#pragma unroll
    for (int w = 0; w < NW; ++w) mg = fmaxf(mg, mlS[w][h][0]);
    float ls = 0.0f, as = 0.0f;
#pragma unroll
    for (int w = 0; w < NW; ++w) {
      const float e = __expf(mlS[w][h][0] - mg);
      ls += e * mlS[w][h][1];
      as += e * accS[w][h][d];
    }
    wsp[h * 130 + d] = as;
    if (d == 0) {
      wsp[h * 130 + 128] = mg;
      wsp[h * 130 + 129] = ls;
    }
  }
}

// ---------------------------------------------------------------------------
// Kernel 2: combine NSPLIT partials per (b,g), normalize, write output.
// ---------------------------------------------------------------------------
__global__ __launch_bounds__(128) void gqa_combine(
    const float* __restrict__ ws, float* __restrict__ out)
{
  const int bg = blockIdx.x;
  const int d  = threadIdx.x;
  const int g  = bg % G_;
  const int b  = bg / G_;

#pragma unroll
  for (int h = 0; h < HG_; ++h) {
    float mg = -1.0e30f;
#pragma unroll
    for (int sp = 0; sp < NSPLIT; ++sp) {
      const float* p = ws + (size_t)(bg * NSPLIT + sp) * WS_STRIDE + h * 130;
      mg = fmaxf(mg, p[128]);
    }
    float ls = 0.0f, as = 0.0f;
#pragma unroll
    for (int sp = 0; sp < NSPLIT; ++sp) {
      const float* p = ws + (size_t)(bg * NSPLIT + sp) * WS_STRIDE + h * 130;
      const float e = __expf(p[128] - mg);
      ls += e * p[129];
      as += e * p[d];
    }
    out[((size_t)b * H_ + g * HG_ + h) * D_ + d] = as / fmaxf(ls, 1e-30f);
  }
}

extern "C" void kernel_launch(void* const* d_in, const int* in_sizes, int n_in,
                              void* d_out, int out_size, void* d_ws, size_t ws_size,
                              hipStream_t stream) {
  const float* Q = (const float*)d_in[0];
  const float* K = (const float*)d_in[1];
  const float* V = (const float*)d_in[2];
  const unsigned char* mask = (const unsigned char*)d_in[3];
  float* out = (float*)d_out;
  float* ws  = (float*)d_ws;  // needs B*G*NSPLIT*520 floats ~ 4.3 MB

  gqa_partial<<<B_ * G_ * NSPLIT, NT_, 0, stream>>>(Q, K, V, mask, ws);
  gqa_combine<<<B_ * G_, 128, 0, stream>>>(ws, out);
}